// Attention_18631568130091
// MI455X (gfx1250) — compile-verified
//
#include <hip/hip_runtime.h>

typedef __attribute__((ext_vector_type(2))) float v2f;
typedef __attribute__((ext_vector_type(8))) float v8f;

#define Bx 8
#define Hx 16
#define Kx 2048
#define Dx 128
#define EPSc 1e-8f
#define CHUNKS 8           // blocks per (b,h)
#define WAVES 8            // waves per block (256 threads, wave32)
#define TILES_PER_WAVE (Kx / (CHUNKS * WAVES * 16))  // = 2

__global__ __launch_bounds__(256) void cosattn_kernel(
    const float* __restrict__ query, const float* __restrict__ key,
    const float* __restrict__ value, const int* __restrict__ mask,
    float* __restrict__ out, float* __restrict__ pattn)
{
    __shared__ float sq[Dx];
    const int tid  = threadIdx.x;
    const int bh   = blockIdx.x / CHUNKS;   // (b*H + h)
    const int chnk = blockIdx.x % CHUNKS;
    const int b    = bh / Hx;
    const int wave = tid >> 5;
    const int lane = tid & 31;

    // Stage q into LDS (512 B)
    if (tid < Dx) sq[tid] = query[bh * Dx + tid];
    __syncthreads();

    // ||q|| : each lane sums 4 elements, then 5-step shfl_xor reduce (wave32)
    float4 q4 = ((const float4*)sq)[lane];
    float qss = q4.x*q4.x + q4.y*q4.y + q4.z*q4.z + q4.w*q4.w;
    qss += __shfl_xor(qss, 16, 32);
    qss += __shfl_xor(qss,  8, 32);
    qss += __shfl_xor(qss,  4, 32);
    qss += __shfl_xor(qss,  2, 32);
    qss += __shfl_xor(qss,  1, 32);
    const float qn = fmaxf(sqrtf(qss), EPSc);

    // A-matrix 16x4 f32 layout: lane L -> row (L&15), K-pair ((L>>4)*2)
    const int r  = lane & 15;
    const int kk = (lane >> 4) << 1;

    for (int t = 0; t < TILES_PER_WAVE; ++t) {
        const int tile  = chnk * (WAVES * TILES_PER_WAVE) + wave * TILES_PER_WAVE + t;
        const int kbase = tile * 16;
        const float* krow = key + ((size_t)bh * Kx + kbase + r) * Dx;

        v8f acc = {0.f, 0.f, 0.f, 0.f, 0.f, 0.f, 0.f, 0.f};
        float ssq = 0.f;
#pragma unroll
        for (int d0 = 0; d0 < Dx; d0 += 4) {
            float2 a = *(const float2*)(krow + d0 + kk);
            v2f av; av[0] = a.x;          av[1] = a.y;
            v2f bv; bv[0] = sq[d0 + kk];  bv[1] = sq[d0 + kk + 1];
            // D[m,n] += sum_k A[m,k] * q[d0+k]  (every column identical)
            acc = __builtin_amdgcn_wmma_f32_16x16x4_f32(
                false, av, false, bv, (short)0, acc, false, false);
            ssq = fmaf(a.x, a.x, ssq);
            ssq = fmaf(a.y, a.y, ssq);
        }
        // combine the two half-row partial sums of squares (lane L <-> L^16)
        ssq += __shfl_xor(ssq, 16, 32);   // lane L now holds full ||k_{L&15}||^2

        float pmine = 0.f;
#pragma unroll
        for (int r2 = 0; r2 < 16; ++r2) {
            // C/D layout: rows 0-7 in lanes 0-15 (vgpr=r2), rows 8-15 in lanes 16-31 (vgpr=r2-8)
            float dotr = __shfl(acc[r2 & 7], (r2 < 8) ? 0 : 16, 32);
            float ssr  = __shfl(ssq, r2, 32);
            const int kidx = kbase + r2;
            const int m = mask[b * Kx + kidx];

            float kn = fmaxf(sqrtf(ssr), EPSc);
            float s  = fabsf(dotr / (qn * kn));
            if (m == 0) s = -1e9f;
            float p = expf(s);
            if (lane == r2) pmine = p;

            // out[k, :] = p * value[k, :]  — coalesced float4 stream
            const float4* vrow = (const float4*)(value + ((size_t)bh * Kx + kidx) * Dx);
            float4*       orow = (float4*)(out  + ((size_t)bh * Kx + kidx) * Dx);
            float4 vv = vrow[lane];
            float4 ov;
            ov.x = vv.x * p; ov.y = vv.y * p; ov.z = vv.z * p; ov.w = vv.w * p;
            orow[lane] = ov;
        }
        if (lane < 16) pattn[(size_t)bh * Kx + kbase + lane] = pmine;
    }
}

extern "C" void kernel_launch(void* const* d_in, const int* in_sizes, int n_in,
                              void* d_out, int out_size, void* d_ws, size_t ws_size,
                              hipStream_t stream) {
    (void)in_sizes; (void)n_in; (void)d_ws; (void)ws_size; (void)out_size;
    const float* query = (const float*)d_in[0];
    const float* key   = (const float*)d_in[1];
    const float* value = (const float*)d_in[2];
    const int*   mask  = (const int*)d_in[3];
    float* out   = (float*)d_out;
    float* pattn = out + (size_t)Bx * Hx * Kx * Dx;   // tuple output: (out, p_attn) flat

    dim3 grid(Bx * Hx * CHUNKS);   // 1024 blocks
    dim3 block(256);               // 8 wave32s
    hipLaunchKernelGGL(cosattn_kernel, grid, block, 0, stream,
                       query, key, value, mask, out, pattn);
}